// DynamicRouting_52905407152297
// MI455X (gfx1250) — compile-verified
//
#include <hip/hip_runtime.h>
#include <hip/hip_bf16.h>
#include <math.h>

// ---------------- problem constants (from reference) ----------------
// C=1024 capsule rows, K=64 input capsules, H=512 features, 3 routing iters.
#define NC 1024
#define NK 64
#define NH 512
#define NITER 3

// padded LDS pitches (floats): 516 % 64 == 4 -> conflict-free ds_load_b64
#define ENCP 516
#define XHP  516

// LDS partition (float indices)
#define ENC_OFF 0
#define XH_OFF  (NK * ENCP)                 // 33024
#define CV_OFF  (XH_OFF + NK * XHP)         // 66048
#define BV_OFF  (CV_OFF + NH)               // 66560
#define DV_OFF  (BV_OFF + NK)               // 66624
#define RED_OFF (DV_OFF + NK)               // 66688
#define SMEM_FLOATS (RED_OFF + 8)           // 66696
#define SMEM_BYTES  (SMEM_FLOATS * 4)       // 266784 B  (< 320 KB / WGP)

typedef float v2f __attribute__((ext_vector_type(2)));
typedef float v8f __attribute__((ext_vector_type(8)));

__device__ __forceinline__ float wave_reduce_add(float x) {
    #pragma unroll
    for (int off = 16; off > 0; off >>= 1)
        x += __shfl_down(x, off, 32);
    return x;
}

// Async global -> LDS copy of 16 bytes per lane (ASYNCcnt-tracked).
// GVS addressing: mem = SGPR64 base + signed VGPR32 offset.
__device__ __forceinline__ void async_g2l_b128(unsigned lds_byte_off,
                                               const void* gbase,
                                               int gbyte_off) {
    asm volatile("global_load_async_to_lds_b128 %0, %1, %2"
                 :: "v"(lds_byte_off), "v"(gbyte_off), "s"(gbase)
                 : "memory");
}

__device__ __forceinline__ void wait_async0() {
    asm volatile("s_wait_asynccnt 0" ::: "memory");
}

__global__ __launch_bounds__(256)
void DynamicRouting_52905407152297_kernel(const float* __restrict__ enc,   // [C,K,H]
                                          const float* __restrict__ W,    // [H,H] row-major (W[g][h])
                                          float* __restrict__ out) {      // [C,H]
    extern __shared__ float smem[];
    float* encLds = smem + ENC_OFF;   // [NK][ENCP]
    float* xhat   = smem + XH_OFF;    // [NK][XHP]
    float* cvec   = smem + CV_OFF;    // [NH]
    float* bvec   = smem + BV_OFF;    // [NK]
    float* dvec   = smem + DV_OFF;    // [NK]
    float* red    = smem + RED_OFF;   // [8]

    const int tid  = threadIdx.x;
    const int wave = tid >> 5;
    const int lane = tid & 31;
    const int lrow = lane & 15;           // M (A) / N (B,D) index within tile
    const int lkof = (lane >> 4) * 2;     // K offset selected by lane[4]
    const int cIdx = blockIdx.x;

    // ---------------- stage enc[cIdx] tile (64x512 f32) into LDS, async ----
    {
        const char* encG = (const char*)(enc + (size_t)cIdx * NK * NH);
        #pragma unroll 4
        for (int i = tid; i < (NK * NH) / 4; i += 256) {   // 8192 x 16B chunks
            int e   = i << 2;                              // float index
            int row = e >> 9;                              // /512
            int col = e & (NH - 1);
            unsigned ldsOff = (unsigned)((ENC_OFF + row * ENCP + col) * 4);
            async_g2l_b128(ldsOff, encG, e * 4);
        }
        if (tid < NK) bvec[tid] = 0.0f;                    // routing logits
        wait_async0();
        __syncthreads();
    }

    // ---------------- GEMM: xhat[k][g] = sum_h enc[k][h] * W[g][h] ---------
    // Wave w owns N-tiles n = w + 8*t (t=0..3) and all 4 M-strips (K rows).
    // f32 WMMA 16x16x4, f32 accumulate -> bit-faithful fp32 math.
    {
        v8f acc[4][4];
        #pragma unroll
        for (int m = 0; m < 4; ++m)
            #pragma unroll
            for (int t = 0; t < 4; ++t)
                acc[m][t] = (v8f){0,0,0,0,0,0,0,0};

        for (int h = 0; h < NH; h += 4) {
            v2f afr[4], bfr[4];
            #pragma unroll
            for (int m = 0; m < 4; ++m) {
                // A frag: lane -> M = m*16+lrow ; VGPR j -> K = h + lkof + j
                const float* p = &encLds[(m * 16 + lrow) * ENCP + h + lkof];
                afr[m] = *(const v2f*)p;                    // ds_load_b64
            }
            #pragma unroll
            for (int t = 0; t < 4; ++t) {
                // B frag: lane -> N = gcol ; VGPR j -> K = h + lkof + j
                int gcol = (wave + 8 * t) * 16 + lrow;
                const float* p = &W[(size_t)gcol * NH + h + lkof];
                bfr[t] = *(const v2f*)p;                    // global_load_b64 (L2 hot)
            }
            #pragma unroll
            for (int m = 0; m < 4; ++m)
                #pragma unroll
                for (int t = 0; t < 4; ++t)
                    acc[m][t] = __builtin_amdgcn_wmma_f32_16x16x4_f32(
                        false, afr[m], false, bfr[t],
                        (short)0, acc[m][t], false, false);
        }

        // D layout -> LDS xhat: lane -> N=lrow(+tile), VGPR j -> M = j + 8*lane[4]
        #pragma unroll
        for (int m = 0; m < 4; ++m)
            #pragma unroll
            for (int t = 0; t < 4; ++t) {
                int colBase = (wave + 8 * t) * 16 + lrow;
                int rowBase = m * 16 + ((lane >> 4) << 3);
                #pragma unroll
                for (int j = 0; j < 8; ++j)
                    xhat[(rowBase + j) * XHP + colBase] = acc[m][t][j];
            }
        __syncthreads();
    }

    // ---------------- dynamic routing, entirely in LDS ---------------------
    const int h0 = tid, h1 = tid + 256;
    for (int it = 0; it < NITER; ++it) {
        // softmax over K=64 (redundant per thread; LDS broadcast reads)
        float bmax = -INFINITY;
        #pragma unroll 8
        for (int k = 0; k < NK; ++k) bmax = fmaxf(bmax, bvec[k]);
        float ssum = 0.0f;
        #pragma unroll 8
        for (int k = 0; k < NK; ++k) ssum += __expf(bvec[k] - bmax);
        float inv = 1.0f / ssum;
        if (tid < NK) dvec[tid] = __expf(bvec[tid] - bmax) * inv;
        __syncthreads();

        // c_hat[h] = sum_k d[k] * xhat[k][h]   (each thread: 2 h values)
        float ch0 = 0.0f, ch1 = 0.0f;
        #pragma unroll 8
        for (int k = 0; k < NK; ++k) {
            float dk = dvec[k];
            ch0 = fmaf(dk, xhat[k * XHP + h0], ch0);
            ch1 = fmaf(dk, xhat[k * XHP + h1], ch1);
        }

        // squash: norm = sum_h c_hat^2 (block reduce over 8 waves)
        float part = wave_reduce_add(ch0 * ch0 + ch1 * ch1);
        if (lane == 0) red[wave] = part;
        __syncthreads();
        float norm = 0.0f;
        #pragma unroll
        for (int w = 0; w < 8; ++w) norm += red[w];
        float scale = (norm / (1.0f + norm)) * rsqrtf(fmaxf(norm, 1e-30f));
        float c0 = ch0 * scale, c1 = ch1 * scale;

        if (it == NITER - 1) {
            float* o = out + (size_t)cIdx * NH;
            o[h0] = c0; o[h1] = c1;
            return;
        }
        cvec[h0] = c0; cvec[h1] = c1;
        __syncthreads();

        // b[k] += sum_h xhat[k][h] * c[h]   (8 k per wave, lane-strided h)
        #pragma unroll
        for (int kk = 0; kk < 8; ++kk) {
            int k = wave * 8 + kk;
            float p = 0.0f;
            #pragma unroll 4
            for (int h = lane; h < NH; h += 32)
                p = fmaf(xhat[k * XHP + h], cvec[h], p);
            p = wave_reduce_add(p);
            if (lane == 0) bvec[k] += p;
        }
        __syncthreads();
    }
}

extern "C" void kernel_launch(void* const* d_in, const int* in_sizes, int n_in,
                              void* d_out, int out_size, void* d_ws, size_t ws_size,
                              hipStream_t stream) {
    (void)in_sizes; (void)n_in; (void)out_size; (void)d_ws; (void)ws_size;
    const float* enc = (const float*)d_in[0];   // [1024,64,512] f32
    const float* W   = (const float*)d_in[1];   // [512,512] f32
    float* out       = (float*)d_out;           // [1024,512] f32

    hipFuncSetAttribute((const void*)DynamicRouting_52905407152297_kernel,
                        hipFuncAttributeMaxDynamicSharedMemorySize, SMEM_BYTES);

    DynamicRouting_52905407152297_kernel<<<NC, 256, SMEM_BYTES, stream>>>(enc, W, out);
}